// InteractionModule_70300024701660
// MI455X (gfx1250) — compile-verified
//
#include <hip/hip_runtime.h>
#include <math.h>

#define NVEC  1024
#define BATCH 64
#define HDIM  64
#define NWL   10
#define NHOP  11

// chord offsets: diag + (r-1+2^t)%N for t=0..10 ; t=0 duplicates the diagonal.
__device__ __constant__ int c_off[NHOP] = {0, 1, 3, 7, 15, 31, 63, 127, 255, 511, 1023};

typedef float v2f __attribute__((ext_vector_type(2)));
typedef float v8f __attribute__((ext_vector_type(8)));

// ---------------------------------------------------------------------------
// One scan step:  V'[b,n] = sum_k ( h[b,n,:] . w2[:,c_k] + b2[c_k] ) * V[b,c_k]
// h[b,n,:] = gelu_exact(data[b,n] * w1 + b1)
// Block: (b, 256-row chunk of n). V[b,:] staged in LDS (4KB).
// ---------------------------------------------------------------------------
__global__ __launch_bounds__(256) void chord_step_kernel(
    const float* __restrict__ data,   // B*N
    const float* __restrict__ w1,     // H (layer slice, D==1)
    const float* __restrict__ b1,     // H
    const float* __restrict__ w2,     // H*N row-major
    const float* __restrict__ b2,     // N
    const float* __restrict__ Vin,    // B*N
    float* __restrict__ Vout)         // B*N
{
    __shared__ float sV[NVEC];
    __shared__ float sW1[HDIM];
    __shared__ float sB1[HDIM];

    const int tid = threadIdx.x;
    const int b   = blockIdx.x;
    const int n   = blockIdx.y * 256 + tid;

    for (int i = tid; i < NVEC; i += 256) sV[i] = Vin[b * NVEC + i];
    if (tid < HDIM) { sW1[tid] = w1[tid]; sB1[tid] = b1[tid]; }
    __syncthreads();

    int   ck[NHOP];
    float vc[NHOP];
    float acc = 0.0f;
#pragma unroll
    for (int k = 0; k < NHOP; ++k) {
        const int c = (n + c_off[k]) & (NVEC - 1);
        ck[k] = c;
        const float v = sV[c];
        vc[k] = v;
        acc = fmaf(v, b2[c], acc);      // bias term of the masked row
    }

    const float x = data[b * NVEC + n];
#pragma unroll 4
    for (int hh = 0; hh < HDIM; ++hh) {
        const float pre = fmaf(x, sW1[hh], sB1[hh]);
        // exact gelu: 0.5*x*(1+erf(x/sqrt(2)))
        const float hv = 0.5f * pre * (1.0f + erff(pre * 0.70710678118654752440f));
        const float* __restrict__ w2row = w2 + hh * NVEC;   // L2-resident (256KB)
        float g = 0.0f;
#pragma unroll
        for (int k = 0; k < NHOP; ++k) g = fmaf(vc[k], w2row[ck[k]], g);
        acc = fmaf(hv, g, acc);
    }

    Vout[b * NVEC + n] = acc;
}

// ---------------------------------------------------------------------------
// Final projection: out(64x2) = V(64x1024) @ Wf(1024x2) + bf
// fp32 WMMA: V_WMMA_F32_16X16X4_F32, one 16-row tile per wave, 4 waves.
// A 16x4 f32 layout: lanes 0-15 M=0..15, VGPR0={K0|K2}, VGPR1={K1|K3}.
// B 4x16 f32 layout: VGPR0 = row K0 (lanes 0-15) / K2 (lanes 16-31), VGPR1 = K1/K3.
// C/D 16x16 f32: VGPR j holds M=j (lanes 0-15) / M=j+8 (lanes 16-31), N=lane&15.
// ---------------------------------------------------------------------------
__global__ __launch_bounds__(128) void final_proj_wmma_kernel(
    const float* __restrict__ V,      // 64 x 1024
    const float* __restrict__ Wf,     // 1024 x 2
    const float* __restrict__ bf,     // 2
    float* __restrict__ out)          // 64 x 2
{
    __shared__ float sWf[NVEC * 2];
    const int tid = threadIdx.x;
    for (int i = tid; i < NVEC * 2; i += 128) sWf[i] = Wf[i];
    __syncthreads();

    const int lane     = tid & 31;
    const int wave     = tid >> 5;
    const int row_base = wave * 16;
    const int m        = lane & 15;          // A row / B-D column
    const int hiHalf   = (lane >> 4) & 1;    // 0 -> K pair {0,1}, 1 -> {2,3}
    const int col      = m;
    const int colSafe  = (col < 2) ? col : 1;

    v8f acc = {0.f, 0.f, 0.f, 0.f, 0.f, 0.f, 0.f, 0.f};

    for (int k0 = 0; k0 < NVEC; k0 += 4) {
        const int kk = k0 + hiHalf * 2;
        v2f va, vb;
        va.x = V[(row_base + m) * NVEC + kk];
        va.y = V[(row_base + m) * NVEC + kk + 1];
        const float b0 = sWf[kk * 2 + colSafe];
        const float b1v = sWf[(kk + 1) * 2 + colSafe];
        vb.x = (col < 2) ? b0  : 0.0f;       // selects, not branches: EXEC stays full
        vb.y = (col < 2) ? b1v : 0.0f;
        // 8 args: (neg_a, A, neg_b, B, c_mod, C, reuse_a, reuse_b)
        acc = __builtin_amdgcn_wmma_f32_16x16x4_f32(false, va, false, vb,
                                                    (short)0, acc, false, false);
    }

#pragma unroll
    for (int j = 0; j < 8; ++j) {
        const int M = j + hiHalf * 8;
        if (col < 2) out[(row_base + M) * 2 + col] = acc[j] + bf[col];
    }
}

// ---------------------------------------------------------------------------
extern "C" void kernel_launch(void* const* d_in, const int* in_sizes, int n_in,
                              void* d_out, int out_size, void* d_ws, size_t ws_size,
                              hipStream_t stream)
{
    const float* data  = (const float*)d_in[0];   // (64,1024,1)
    const float* f_w1  = (const float*)d_in[1];   // (10,1,64)
    const float* f_b1  = (const float*)d_in[2];   // (10,64)
    const float* f_w2  = (const float*)d_in[3];   // (10,64,1024)
    const float* f_b2  = (const float*)d_in[4];   // (10,1024)
    const float* fin_w = (const float*)d_in[5];   // (1024,2)
    const float* fin_b = (const float*)d_in[6];   // (2,)
    // d_in[7] = mask, implied by hardcoded chord offsets
    float* outp = (float*)d_out;                  // (64,2)

    float* bufA = (float*)d_ws;                   // B*N fp32 ping
    float* bufB = bufA + BATCH * NVEC;            // B*N fp32 pong

    const dim3 grid(BATCH, NVEC / 256);
    const float* cur = data;                      // scan carry starts at data
    float* nxt = bufA;
    for (int i = 0; i < NWL; ++i) {
        const int layer = NWL - 1 - i;            // params are reversed in the scan
        chord_step_kernel<<<grid, 256, 0, stream>>>(
            data,
            f_w1 + layer * HDIM,
            f_b1 + layer * HDIM,
            f_w2 + (size_t)layer * HDIM * NVEC,
            f_b2 + layer * NVEC,
            cur, nxt);
        cur = nxt;
        nxt = (nxt == bufA) ? bufB : bufA;
    }

    final_proj_wmma_kernel<<<1, 128, 0, stream>>>(cur, fin_w, fin_b, outp);
}